// MGD_loss_full_eye_85323820302378
// MI455X (gfx1250) — compile-verified
//
#include <hip/hip_runtime.h>
#include <hip/hip_bf16.h>
#include <math.h>

// Problem constants (match reference)
constexpr int B_   = 128;
constexpr int Q_   = 24;
constexpr int N_   = 4000;
constexpr int RN_  = 30;
constexpr int RQ_  = 12;
constexpr int R_   = 360;          // RQ*RN
constexpr int CHUNKS = 4;          // N split for parallelism
constexpr int CN_  = N_ / CHUNKS;  // 1000
constexpr int KB_  = 100;          // K staging block in LDS (multiple of 4)

typedef __attribute__((ext_vector_type(2))) float v2f;
typedef __attribute__((ext_vector_type(8))) float v8f;

// ---------------------------------------------------------------------------
// Kernel 1: per (batch, N-chunk) partial GEMM  M_b += x_b(24xK) @ LnScaled(Kx30)
// using V_WMMA_F32_16X16X4_F32, plus partial ||x||^2.
// 128 threads = 4 waves; wave w owns output tile (mt = w&1 rows, nt = w>>1 cols).
// Global streaming is done as float4 (GLOBAL_LOAD_B128) for bandwidth.
// ---------------------------------------------------------------------------
__global__ __launch_bounds__(128)
void mgd_partial_gemm_kernel(const float* __restrict__ eps,
                             const float* __restrict__ y,
                             const float* __restrict__ Ln,
                             float* __restrict__ pM,   // [B][CHUNKS][32][32]
                             float* __restrict__ pSS)  // [B][CHUNKS]
{
    __shared__ float xs[32][KB_];   // masked x rows (rows 24..31 stay zero)
    __shared__ float ls[KB_][32];   // scaled Ln, cols 30..31 stay zero
    __shared__ float red[128];

    const int blk   = blockIdx.x;
    const int b     = blk / CHUNKS;
    const int ch    = blk % CHUNKS;
    const int nbase = ch * CN_;
    const int tid   = threadIdx.x;
    const int lane  = tid & 31;
    const int w     = tid >> 5;
    const int mt    = w & 1;
    const int nt    = w >> 1;
    const int lrow  = lane & 15;
    const int hi    = lane >> 4;

    // one-time zeroing of pads (never rewritten afterwards)
    for (int idx = tid; idx < 8 * KB_; idx += 128)
        xs[24 + idx / KB_][idx % KB_] = 0.0f;          // xs rows 24..31
    for (int idx = tid; idx < 2 * KB_; idx += 128)
        ls[idx >> 1][30 + (idx & 1)] = 0.0f;           // ls cols 30,31

    const float scaleN = 0.18257418583505536f; // 1/sqrt(30)
    v8f acc = {};
    float ss = 0.0f;

    for (int kb = 0; kb < CN_; kb += KB_) {
        __syncthreads(); // protect LDS reuse vs. previous stage's WMMA reads

        // stage masked x: rows 0..23, cols kb..kb+KB_, float4 loads/stores
        for (int idx = tid; idx < Q_ * (KB_ / 4); idx += 128) {
            const int q  = idx / (KB_ / 4);
            const int c4 = (idx % (KB_ / 4)) * 4;
            const size_t g = (size_t)(b * Q_ + q) * N_ + (nbase + kb + c4);
            const float4 e4 = *reinterpret_cast<const float4*>(eps + g);
            const float4 y4 = *reinterpret_cast<const float4*>(y + g);
            float4 v;
            v.x = (y4.x != 0.0f) ? e4.x : 0.0f;
            v.y = (y4.y != 0.0f) ? e4.y : 0.0f;
            v.z = (y4.z != 0.0f) ? e4.z : 0.0f;
            v.w = (y4.w != 0.0f) ? e4.w : 0.0f;
            ss += v.x * v.x + v.y * v.y + v.z * v.z + v.w * v.w;
            *reinterpret_cast<float4*>(&xs[q][c4]) = v;
        }
        // stage scaled Ln chunk: contiguous 3000 floats, float4 loads,
        // scalar scatter into padded [k][32] layout
        for (int idx = tid; idx < (KB_ * RN_) / 4; idx += 128) {
            const float4 l4 = *reinterpret_cast<const float4*>(
                Ln + (size_t)(nbase + kb) * RN_ + idx * 4);
            const int e = idx * 4;
            ls[(e + 0) / RN_][(e + 0) % RN_] = l4.x * scaleN;
            ls[(e + 1) / RN_][(e + 1) % RN_] = l4.y * scaleN;
            ls[(e + 2) / RN_][(e + 2) % RN_] = l4.z * scaleN;
            ls[(e + 3) / RN_][(e + 3) % RN_] = l4.w * scaleN;
        }
        // prefetch next stage (gfx1250 global_prefetch_b8)
        if (kb + KB_ < CN_ && tid < Q_) {
            __builtin_prefetch(&eps[(size_t)(b * Q_ + tid) * N_ + nbase + kb + KB_], 0, 0);
            __builtin_prefetch(&y  [(size_t)(b * Q_ + tid) * N_ + nbase + kb + KB_], 0, 0);
        }
        __syncthreads();

        // WMMA over this K block: D(16x16) += A(16x4) * B(4x16)
        // A layout (f32 16x4): lanes 0-15 hold K=0(v0),K=1(v1); lanes 16-31 K=2,K=3.
        // B layout (f32 4x16): v0 = rows K=0 / K=2 split across lane halves, v1 = K=1 / K=3.
        for (int k0 = 0; k0 < KB_; k0 += 4) {
            v2f a, bb;
            a.x  = xs[mt * 16 + lrow][k0 + 2 * hi + 0];
            a.y  = xs[mt * 16 + lrow][k0 + 2 * hi + 1];
            bb.x = ls[k0 + 2 * hi + 0][nt * 16 + lrow];
            bb.y = ls[k0 + 2 * hi + 1][nt * 16 + lrow];
            acc = __builtin_amdgcn_wmma_f32_16x16x4_f32(
                false, a, false, bb, (short)0, acc, false, false);
        }
    }

    // write partial tile: C/D layout = 8 VGPRs, lanes 0-15 rows 0..7, lanes 16-31 rows 8..15
    {
        float* out = pM + (size_t)(b * CHUNKS + ch) * 1024;
        #pragma unroll
        for (int r = 0; r < 8; ++r) {
            const int row = mt * 16 + hi * 8 + r;
            const int col = nt * 16 + lrow;
            out[row * 32 + col] = acc[r];
        }
    }

    // block-reduce partial sum of squares
    red[tid] = ss;
    __syncthreads();
    for (int s = 64; s > 0; s >>= 1) {
        if (tid < s) red[tid] += red[tid + s];
        __syncthreads();
    }
    if (tid == 0) pSS[b * CHUNKS + ch] = red[0];
}

// ---------------------------------------------------------------------------
// Kernel 2 (grid=1, 512 threads = 16 waves): everything tiny.
//   A) c from sigma; An = Ln^T Ln / 30 ; Aq = LqS^T LqS ; per-batch sumsq
//   B) z[b] = LqS^T M_b  (reduce chunk partials on the fly)
//   C) cap = I + kron(Aq, An)/c  in global scratch (518 KB)
//   D) in-place blocked-parallel Cholesky of cap
//   E) wave-cooperative forward substitution per batch (u lane-sharded in regs),
//      maha via Woodbury, mean loss -> out[0]
// ---------------------------------------------------------------------------
__global__ __launch_bounds__(512)
void mgd_finalize_kernel(const float* __restrict__ Ln,
                         const float* __restrict__ Lq,
                         const float* __restrict__ sigma,
                         const float* __restrict__ pM,
                         const float* __restrict__ pSS,
                         float* __restrict__ Z,     // [B][R]
                         float* __restrict__ CAP,   // [R][R]
                         float* __restrict__ out)
{
    __shared__ float An[RN_][RN_];
    __shared__ float Aq[RQ_][RQ_];
    __shared__ float LqS[Q_][RQ_];
    __shared__ float ssq[B_];
    __shared__ float lossv[B_];
    __shared__ float s_c, s_invc, s_logdet, s_dk;

    const int tid  = threadIdx.x;
    const int lane = tid & 31;
    const int w    = tid >> 5;
    const int NT   = 512;

    if (tid == 0) {
        const float diag_bias = logf(expm1f(1.0f));   // softplus_inv(sigma_init^2)
        const float xsp = sigma[0] + diag_bias;
        const float c = ((xsp > 20.0f) ? xsp : log1pf(expf(xsp))) + 1.0e-6f;
        s_c = c;
        s_invc = 1.0f / c;
        s_logdet = (float)(Q_ * N_) * logf(c);        // D * log(c) part of logdet
    }
    const float scaleQ = 0.2886751345948129f; // 1/sqrt(12)
    for (int idx = tid; idx < Q_ * RQ_; idx += NT)
        LqS[idx / RQ_][idx % RQ_] = Lq[idx] * scaleQ;
    __syncthreads();

    // An = (Ln^T Ln)/30  (= LnS^T LnS)
    for (int idx = tid; idx < RN_ * RN_; idx += NT) {
        const int i = idx / RN_, j = idx % RN_;
        float s = 0.0f;
        for (int n = 0; n < N_; ++n)
            s += Ln[(size_t)n * RN_ + i] * Ln[(size_t)n * RN_ + j];
        An[i][j] = s * (1.0f / 30.0f);
    }
    // Aq = LqS^T LqS
    for (int idx = tid; idx < RQ_ * RQ_; idx += NT) {
        const int i = idx / RQ_, j = idx % RQ_;
        float s = 0.0f;
        for (int q = 0; q < Q_; ++q) s += LqS[q][i] * LqS[q][j];
        Aq[i][j] = s;
    }
    // per-batch sumsq
    for (int b = tid; b < B_; b += NT) {
        float s = 0.0f;
        for (int c2 = 0; c2 < CHUNKS; ++c2) s += pSS[b * CHUNKS + c2];
        ssq[b] = s;
    }
    // z[b][i*RN+j] = sum_q LqS[q][i] * (sum_ch pM[b][ch][q][j])
    for (int idx = tid; idx < B_ * RQ_ * RN_; idx += NT) {
        const int b   = idx / (RQ_ * RN_);
        const int rem = idx % (RQ_ * RN_);
        const int i   = rem / RN_, j = rem % RN_;
        float s = 0.0f;
        for (int q = 0; q < Q_; ++q) {
            float m = 0.0f;
            for (int c2 = 0; c2 < CHUNKS; ++c2)
                m += pM[(size_t)(b * CHUNKS + c2) * 1024 + q * 32 + j];
            s += LqS[q][i] * m;
        }
        Z[(size_t)b * R_ + rem] = s;
    }
    __syncthreads();

    // cap = I + kron(Aq, An)/c
    {
        const float invc = s_invc;
        for (int idx = tid; idx < R_ * R_; idx += NT) {
            const int r = idx / R_, s2 = idx % R_;
            float v = Aq[r / RN_][s2 / RN_] * An[r % RN_][s2 % RN_] * invc;
            if (r == s2) v += 1.0f;
            CAP[idx] = v;
        }
    }
    __syncthreads();

    // Right-looking Cholesky (lower triangle), in global scratch (L2-resident)
    for (int k = 0; k < R_; ++k) {
        if (tid == 0) {
            const float d = sqrtf(CAP[k * R_ + k]);
            CAP[k * R_ + k] = d;
            s_dk = d;
            s_logdet += 2.0f * logf(d);
        }
        __syncthreads();
        const float inv_d = 1.0f / s_dk;
        for (int i = k + 1 + tid; i < R_; i += NT)
            CAP[i * R_ + k] *= inv_d;
        __syncthreads();
        const int m = R_ - k - 1;
        const int npairs = m * (m + 1) / 2;
        for (int p = tid; p < npairs; p += NT) {
            int i2 = (int)((sqrtf(8.0f * (float)p + 1.0f) - 1.0f) * 0.5f);
            while ((i2 + 1) * (i2 + 2) / 2 <= p) ++i2;
            while (i2 * (i2 + 1) / 2 > p) --i2;
            const int j2 = p - i2 * (i2 + 1) / 2;
            const int i = k + 1 + i2, j = k + 1 + j2;
            CAP[i * R_ + j] -= CAP[i * R_ + k] * CAP[j * R_ + k];
        }
        __syncthreads();
    }

    // Wave-cooperative forward substitution: u = L^{-1} z_b, u lane-sharded
    const float invc   = s_invc;
    const float logdet = s_logdet;
    const float Dlog2pi = (float)(Q_ * N_) * 1.8378770664093453f; // D*log(2*pi)
    float ureg[12];
    for (int g = 0; g < B_ / 16; ++g) {
        const int b = w + 16 * g;
        const float* zb = Z + (size_t)b * R_;
        float ssu = 0.0f;
        #pragma unroll
        for (int t = 0; t < 12; ++t) ureg[t] = 0.0f;
        for (int i = 0; i < R_; ++i) {
            float partial = 0.0f;
            #pragma unroll
            for (int t = 0; t < 12; ++t) {
                const int kk = lane + 32 * t;
                if (kk < i) partial += CAP[i * R_ + kk] * ureg[t];
            }
            #pragma unroll
            for (int off = 16; off > 0; off >>= 1)
                partial += __shfl_xor(partial, off, 32);
            const float val = (zb[i] - partial) / CAP[i * R_ + i];
            if (lane == (i & 31)) {
                ureg[i >> 5] = val;
                ssu += val * val;
            }
        }
        #pragma unroll
        for (int off = 16; off > 0; off >>= 1)
            ssu += __shfl_xor(ssu, off, 32);
        if (lane == 0) {
            const float maha = ssq[b] * invc - ssu * invc * invc;
            lossv[b] = 0.5f * (Dlog2pi + logdet + maha);
        }
    }
    __syncthreads();
    if (tid == 0) {
        float s = 0.0f;
        for (int b = 0; b < B_; ++b) s += lossv[b];
        out[0] = s / (float)B_;
    }
}

extern "C" void kernel_launch(void* const* d_in, const int* in_sizes, int n_in,
                              void* d_out, int out_size, void* d_ws, size_t ws_size,
                              hipStream_t stream) {
    const float* eps = (const float*)d_in[0];
    const float* y   = (const float*)d_in[1];
    const float* Ln  = (const float*)d_in[2];
    const float* Lq  = (const float*)d_in[3];
    const float* sg  = (const float*)d_in[4];

    float* ws  = (float*)d_ws;
    float* pM  = ws;                                  // B*CHUNKS*1024 floats
    float* pSS = pM + (size_t)B_ * CHUNKS * 1024;     // B*CHUNKS
    float* Z   = pSS + (size_t)B_ * CHUNKS;           // B*R
    float* CAP = Z + (size_t)B_ * R_;                 // R*R  (total ~2.8 MB)

    mgd_partial_gemm_kernel<<<dim3(B_ * CHUNKS), dim3(128), 0, stream>>>(
        eps, y, Ln, pM, pSS);
    mgd_finalize_kernel<<<dim3(1), dim3(512), 0, stream>>>(
        Ln, Lq, sg, pM, pSS, Z, CAP, (float*)d_out);
}